// AttentionCopyDecoder_6622839571281
// MI455X (gfx1250) — compile-verified
//
#include <hip/hip_runtime.h>
#include <math.h>

// Problem dims
#define Bn     64
#define Tn     400
#define H2n    1024
#define En     256
#define Vn     50000
#define G3n    3072     // 3*H2
#define KOUTn  2304     // 2*H2 + E
#define KCOMBn 1280     // H2 + E

typedef float v2f __attribute__((ext_vector_type(2)));
typedef float v4f __attribute__((ext_vector_type(4)));
typedef float v8f __attribute__((ext_vector_type(8)));

__device__ __forceinline__ float wred_sum(float v) {
#pragma unroll
  for (int o = 16; o; o >>= 1) v += __shfl_down(v, o);
  return v;
}
__device__ __forceinline__ float wred_max(float v) {
#pragma unroll
  for (int o = 16; o; o >>= 1) v = fmaxf(v, __shfl_down(v, o));
  return v;
}
__device__ __forceinline__ float sigf(float x) { return 1.f / (1.f + expf(-x)); }

// -------------------------------------------------------------------------
// K1: per-batch attention scores + softmax + context (fused, one WG per b)
// -------------------------------------------------------------------------
__global__ __launch_bounds__(512) void attn_context_kernel(
    const float* __restrict__ enc, const float* __restrict__ pre_h,
    const float* __restrict__ W_align, const float* __restrict__ b_align,
    float* __restrict__ attn_out, float* __restrict__ attn_ws,
    float* __restrict__ context)
{
  __shared__ __align__(16) float s_wa[H2n];
  __shared__ float s_sc[Tn];
  __shared__ float s_red[16];
  __shared__ float s_val;
  int b = blockIdx.x, tid = threadIdx.x;
  int lane = tid & 31, wave = tid >> 5;

  for (int i = tid; i < H2n; i += 512) s_wa[i] = W_align[i];
  // c_b = dot(pre_hidden[b], W_align[H2:2H2]) + b_align
  float p = 0.f;
  for (int i = tid; i < H2n; i += 512) p += pre_h[b * H2n + i] * W_align[H2n + i];
  p = wred_sum(p);
  if (lane == 0) s_red[wave] = p;
  __syncthreads();
  if (tid == 0) { float s = 0.f; for (int w = 0; w < 16; w++) s += s_red[w]; s_val = s + b_align[0]; }
  __syncthreads();
  float cb = s_val;
  const float* encb = enc + (size_t)b * Tn * H2n;

  // scores: one wave per t, float4 per lane (coalesced 512B per wave step)
  for (int t = wave; t < Tn; t += 16) {
    const float* row = encb + (size_t)t * H2n;
    float acc = 0.f;
    for (int j = lane * 4; j < H2n; j += 128) {
      v4f r = *(const v4f*)(row + j);
      v4f w = *(const v4f*)(s_wa + j);
      acc += r.x * w.x + r.y * w.y + r.z * w.z + r.w * w.w;
    }
    acc = wred_sum(acc);
    if (lane == 0) s_sc[t] = tanhf(acc + cb);
  }
  __syncthreads();

  // softmax over T in LDS
  float m = -3.4e38f;
  for (int t = tid; t < Tn; t += 512) m = fmaxf(m, s_sc[t]);
  m = wred_max(m);
  if (lane == 0) s_red[wave] = m;
  __syncthreads();
  if (tid == 0) { float mm = -3.4e38f; for (int w = 0; w < 16; w++) mm = fmaxf(mm, s_red[w]); s_val = mm; }
  __syncthreads();
  float mx = s_val;
  float ss = 0.f;
  for (int t = tid; t < Tn; t += 512) { float e = expf(s_sc[t] - mx); s_sc[t] = e; ss += e; }
  ss = wred_sum(ss);
  if (lane == 0) s_red[wave] = ss;
  __syncthreads();
  if (tid == 0) { float s2 = 0.f; for (int w = 0; w < 16; w++) s2 += s_red[w]; s_val = s2; }
  __syncthreads();
  float inv = 1.f / s_val;
  for (int t = tid; t < Tn; t += 512) {
    float a = s_sc[t] * inv; s_sc[t] = a;
    attn_out[b * Tn + t] = a; attn_ws[b * Tn + t] = a;
  }
  __syncthreads();

  // context[h] = sum_t attn[t] * enc[b,t,h]; float2 per thread, coalesced per t
  float acc0 = 0.f, acc1 = 0.f;
  int h0 = tid * 2;
  for (int t = 0; t < Tn; t++) {
    float a = s_sc[t];
    v2f r = *(const v2f*)(encb + (size_t)t * H2n + h0);
    acc0 += a * r.x; acc1 += a * r.y;
  }
  v2f cv; cv.x = acc0; cv.y = acc1;
  *(v2f*)(context + b * H2n + h0) = cv;
}

// -------------------------------------------------------------------------
// K2: emb gather + x = relu([emb, context] @ W_comb.T + b_comb)
// -------------------------------------------------------------------------
__global__ __launch_bounds__(256) void comb_kernel(
    const int* __restrict__ ids, const float* __restrict__ emb_table,
    const float* __restrict__ context, const float* __restrict__ W_comb,
    const float* __restrict__ b_comb,
    float* __restrict__ emb_ws, float* __restrict__ x_ws)
{
  __shared__ __align__(16) float s_cat[KCOMBn];
  int b = blockIdx.x, tid = threadIdx.x;
  int id = ids[b];
  if (tid < En) { float e = emb_table[(size_t)id * En + tid]; s_cat[tid] = e; emb_ws[b * En + tid] = e; }
  for (int i = tid; i < H2n; i += 256) s_cat[En + i] = context[b * H2n + i];
  __syncthreads();
  const float* wrow = W_comb + (size_t)tid * KCOMBn;
  float acc = b_comb[tid];
  for (int k = 0; k < KCOMBn; k += 4) {
    v4f w = *(const v4f*)(wrow + k);
    v4f s = *(const v4f*)(s_cat + k);
    acc += w.x * s.x + w.y * s.y + w.z * s.z + w.w * s.w;
  }
  x_ws[b * En + tid] = fmaxf(acc, 0.f);
}

// -------------------------------------------------------------------------
// K3: GRU gates gi = x@W_ih.T + b_ih, gh = h@W_hh.T + b_hh (wave per (r,b))
// -------------------------------------------------------------------------
__global__ __launch_bounds__(256) void gates_kernel(
    const float* __restrict__ W_ih, const float* __restrict__ W_hh,
    const float* __restrict__ b_ih, const float* __restrict__ b_hh,
    const float* __restrict__ x_ws, const float* __restrict__ pre_h,
    float* __restrict__ gi, float* __restrict__ gh)
{
  int r = blockIdx.x;
  int tid = threadIdx.x, lane = tid & 31, wave = tid >> 5;
  const float* wi = W_ih + (size_t)r * En;
  const float* wh = W_hh + (size_t)r * H2n;
  float bi = b_ih[r], bh = b_hh[r];
  for (int b = wave; b < Bn; b += 8) {
    const float* x = x_ws + b * En;
    const float* h = pre_h + b * H2n;
    float a = 0.f, c = 0.f;
    for (int k = lane * 4; k < En; k += 128) {
      v4f w = *(const v4f*)(wi + k);
      v4f xv = *(const v4f*)(x + k);
      a += w.x * xv.x + w.y * xv.y + w.z * xv.z + w.w * xv.w;
    }
    for (int k = lane * 4; k < H2n; k += 128) {
      v4f w = *(const v4f*)(wh + k);
      v4f hv = *(const v4f*)(h + k);
      c += w.x * hv.x + w.y * hv.y + w.z * hv.z + w.w * hv.w;
    }
    a = wred_sum(a); c = wred_sum(c);
    if (lane == 0) { gi[(size_t)b * G3n + r] = a + bi; gh[(size_t)b * G3n + r] = c + bh; }
  }
}

// -------------------------------------------------------------------------
// K4: GRU pointwise + h_new + feat = [emb, context, h_new]
// -------------------------------------------------------------------------
__global__ __launch_bounds__(256) void gru_feat_kernel(
    const float* __restrict__ gi, const float* __restrict__ gh,
    const float* __restrict__ pre_h, const float* __restrict__ context,
    const float* __restrict__ emb_ws,
    float* __restrict__ hidden_out, float* __restrict__ hn_ws,
    float* __restrict__ feat)
{
  int b = blockIdx.x, tid = threadIdx.x;
  const float* gib = gi + (size_t)b * G3n;
  const float* ghb = gh + (size_t)b * G3n;
  for (int h = tid; h < H2n; h += 256) {
    float r = sigf(gib[h] + ghb[h]);
    float z = sigf(gib[H2n + h] + ghb[H2n + h]);
    float n = tanhf(gib[2 * H2n + h] + r * ghb[2 * H2n + h]);
    float hp = pre_h[b * H2n + h];
    float hn = (1.f - z) * n + z * hp;
    hidden_out[b * H2n + h] = hn;
    hn_ws[b * H2n + h] = hn;
    feat[(size_t)b * KOUTn + KCOMBn + h] = hn;
    feat[(size_t)b * KOUTn + En + h] = context[b * H2n + h];
  }
  for (int e = tid; e < En; e += 256) feat[(size_t)b * KOUTn + e] = emb_ws[b * En + e];
}

// -------------------------------------------------------------------------
// WMMA helper: one float4 A/B load per lane feeds TWO 16x16x4 f32 WMMAs.
// Lane mapping: l16 = lane&15 (A: m-index, B: n-index), hi = lane>>4.
// Lane loads float4 at (k + hi*4). WMMA #1 consumes (x,y) -> global k-set
// {k,k+1,k+4,k+5}; WMMA #2 consumes (z,w) -> {k+2,k+3,k+6,k+7}. The
// slot->k mapping is identical for A and B frags, and WMMA's K reduction is
// order-independent, so the union over both WMMAs covers k..k+7 exactly.
// -------------------------------------------------------------------------
__device__ __forceinline__ v8f wmma_k8(v8f c, const float* __restrict__ Arow,
                                       const float* __restrict__ Brow,
                                       int k, int hi)
{
  v4f a4 = *(const v4f*)(Arow + k + hi * 4);
  v4f b4 = *(const v4f*)(Brow + k + hi * 4);
  v2f a0; a0.x = a4.x; a0.y = a4.y;
  v2f b0; b0.x = b4.x; b0.y = b4.y;
  c = __builtin_amdgcn_wmma_f32_16x16x4_f32(false, a0, false, b0, (short)0, c, false, false);
  v2f a1; a1.x = a4.z; a1.y = a4.w;
  v2f b1; b1.x = b4.z; b1.y = b4.w;
  c = __builtin_amdgcn_wmma_f32_16x16x4_f32(false, a1, false, b1, (short)0, c, false, false);
  return c;
}

// -------------------------------------------------------------------------
// K5: logits = feat @ W_out.T + b_out  via V_WMMA_F32_16X16X4_F32
// grid = V/16 n-tiles; 4 waves per WG = 4 m-tiles sharing the B tile via L0/L2
// -------------------------------------------------------------------------
__global__ __launch_bounds__(128) void out_gemm_kernel(
    const float* __restrict__ feat, const float* __restrict__ W_out,
    const float* __restrict__ b_out, float* __restrict__ logits)
{
  int tid = threadIdx.x;
  int lane = tid & 31, wave = tid >> 5;
  int n0 = blockIdx.x * 16;
  int m0 = wave * 16;
  int l16 = lane & 15, hi = lane >> 4;
  const float* Arow = feat + (size_t)(m0 + l16) * KOUTn;
  const float* Brow = W_out + (size_t)(n0 + l16) * KOUTn;
  v8f c = {0.f, 0.f, 0.f, 0.f, 0.f, 0.f, 0.f, 0.f};
  for (int k = 0; k < KOUTn; k += 8)
    c = wmma_k8(c, Arow, Brow, k, hi);
  float bo = b_out[n0 + l16];
#pragma unroll
  for (int j = 0; j < 8; j++) {
    int m = m0 + j + hi * 8;               // C layout: VGPR j -> M = j + 8*(lane>=16)
    logits[(size_t)m * Vn + n0 + l16] = c[j] + bo;
  }
}

// -------------------------------------------------------------------------
// K6: zero copy_logits accumulator
// -------------------------------------------------------------------------
__global__ void zero_kernel(float* __restrict__ p, int n) {
  int i = blockIdx.x * 256 + threadIdx.x;
  if (i < n) p[i] = 0.f;
}

// -------------------------------------------------------------------------
// K7: copy path: C = enc_out @ W_copy.T, fused sigmoid * h_new + reduce over h
// grid = 1600 m-tiles * 16 n-groups; 4 waves = 4 n-tiles each (full K=1024)
// -------------------------------------------------------------------------
__global__ __launch_bounds__(128) void copy_gemm_kernel(
    const float* __restrict__ enc, const float* __restrict__ W_copy,
    const float* __restrict__ b_copy, const float* __restrict__ hn_ws,
    float* __restrict__ copy_logits)
{
  int tid = threadIdx.x, lane = tid & 31, wave = tid >> 5;
  int mtile = blockIdx.x >> 4;          // 0..1599 (16 bt-rows each; 400%16==0 so one b per tile)
  int ngrp  = blockIdx.x & 15;
  int ntile = ngrp * 4 + wave;          // 0..63
  int m0 = mtile * 16;
  int b = m0 / Tn, t0 = m0 % Tn;
  int n0 = ntile * 16;
  int l16 = lane & 15, hi = lane >> 4;
  const float* Arow = enc + (size_t)(m0 + l16) * H2n;
  const float* Brow = W_copy + (size_t)(n0 + l16) * H2n;
  v8f c = {0.f, 0.f, 0.f, 0.f, 0.f, 0.f, 0.f, 0.f};
  for (int k = 0; k < H2n; k += 8)
    c = wmma_k8(c, Arow, Brow, k, hi);
  int ng = n0 + l16;
  float bc = b_copy[ng];
  float hv = hn_ws[b * H2n + ng];
  float vals[8];
#pragma unroll
  for (int j = 0; j < 8; j++) vals[j] = sigf(c[j] + bc) * hv;
  // reduce over n: sum each half-wave's 16 lanes -> lanes 0 (m=j) and 16 (m=j+8)
#pragma unroll
  for (int j = 0; j < 8; j++) {
#pragma unroll
    for (int o = 8; o; o >>= 1) vals[j] += __shfl_down(vals[j], o);
  }
  if (l16 == 0) {
#pragma unroll
    for (int j = 0; j < 8; j++)
      atomicAdd(&copy_logits[b * Tn + t0 + hi * 8 + j], vals[j]);
  }
}

// -------------------------------------------------------------------------
// K8: softmax over V per batch row (3 passes, logits L2-resident, float4)
// -------------------------------------------------------------------------
__global__ __launch_bounds__(256) void gen_softmax_kernel(
    const float* __restrict__ logits, float* __restrict__ out)
{
  __shared__ float s_red[8];
  __shared__ float s_val;
  int b = blockIdx.x, tid = threadIdx.x, lane = tid & 31, wave = tid >> 5;
  const float* lg = logits + (size_t)b * Vn;
  const int V4 = Vn / 4;   // 12500
  float m = -3.4e38f;
  for (int i = tid; i < V4; i += 256) {
    v4f r = *(const v4f*)(lg + i * 4);
    m = fmaxf(m, fmaxf(fmaxf(r.x, r.y), fmaxf(r.z, r.w)));
  }
  m = wred_max(m);
  if (lane == 0) s_red[wave] = m;
  __syncthreads();
  if (tid == 0) { float mm = -3.4e38f; for (int w = 0; w < 8; w++) mm = fmaxf(mm, s_red[w]); s_val = mm; }
  __syncthreads();
  float mx = s_val;
  float ss = 0.f;
  for (int i = tid; i < V4; i += 256) {
    v4f r = *(const v4f*)(lg + i * 4);
    ss += expf(r.x - mx) + expf(r.y - mx) + expf(r.z - mx) + expf(r.w - mx);
  }
  ss = wred_sum(ss);
  if (lane == 0) s_red[wave] = ss;
  __syncthreads();
  if (tid == 0) { float s2 = 0.f; for (int w = 0; w < 8; w++) s2 += s_red[w]; s_val = s2; }
  __syncthreads();
  float inv = 1.f / s_val;
  float* ob = out + (size_t)b * Vn;
  for (int i = tid; i < V4; i += 256) {
    v4f r = *(const v4f*)(lg + i * 4);
    v4f o;
    o.x = expf(r.x - mx) * inv; o.y = expf(r.y - mx) * inv;
    o.z = expf(r.z - mx) * inv; o.w = expf(r.w - mx) * inv;
    *(v4f*)(ob + i * 4) = o;
  }
}

// -------------------------------------------------------------------------
// K9: out += scatter of attn onto source_input positions
// -------------------------------------------------------------------------
__global__ __launch_bounds__(512) void pos_scatter_kernel(
    const int* __restrict__ src, const float* __restrict__ attn_ws,
    float* __restrict__ out)
{
  int b = blockIdx.x, t = threadIdx.x;
  if (t < Tn) atomicAdd(&out[(size_t)b * Vn + src[b * Tn + t]], attn_ws[b * Tn + t]);
}

// -------------------------------------------------------------------------
// K10: copy_w = softmax over T of copy_logits
// -------------------------------------------------------------------------
__global__ __launch_bounds__(512) void copy_softmax_kernel(
    const float* __restrict__ copy_logits, float* __restrict__ copy_w)
{
  __shared__ float s[Tn];
  __shared__ float s_red[16];
  __shared__ float s_val;
  int b = blockIdx.x, tid = threadIdx.x, lane = tid & 31, wave = tid >> 5;
  if (tid < Tn) s[tid] = copy_logits[b * Tn + tid];
  __syncthreads();
  float m = (tid < Tn) ? s[tid] : -3.4e38f;
  m = wred_max(m);
  if (lane == 0) s_red[wave] = m;
  __syncthreads();
  if (tid == 0) { float mm = -3.4e38f; for (int w = 0; w < 16; w++) mm = fmaxf(mm, s_red[w]); s_val = mm; }
  __syncthreads();
  float mx = s_val;
  float e = (tid < Tn) ? expf(s[tid] - mx) : 0.f;
  float ss = wred_sum(e);
  if (lane == 0) s_red[wave] = ss;
  __syncthreads();
  if (tid == 0) { float s2 = 0.f; for (int w = 0; w < 16; w++) s2 += s_red[w]; s_val = s2; }
  __syncthreads();
  if (tid < Tn) copy_w[b * Tn + tid] = e / s_val;
}

// -------------------------------------------------------------------------
extern "C" void kernel_launch(void* const* d_in, const int* in_sizes, int n_in,
                              void* d_out, int out_size, void* d_ws, size_t ws_size,
                              hipStream_t stream)
{
  (void)in_sizes; (void)n_in; (void)out_size; (void)ws_size;
  const int*   input_ids  = (const int*)  d_in[0];
  const float* pre_hidden = (const float*)d_in[1];
  const float* enc_out    = (const float*)d_in[2];
  const int*   source_in  = (const int*)  d_in[3];
  const float* emb_table  = (const float*)d_in[4];
  const float* W_align    = (const float*)d_in[5];
  const float* b_align    = (const float*)d_in[6];
  const float* W_ih       = (const float*)d_in[7];
  const float* W_hh       = (const float*)d_in[8];
  const float* b_ih       = (const float*)d_in[9];
  const float* b_hh       = (const float*)d_in[10];
  const float* W_comb     = (const float*)d_in[11];
  const float* b_comb     = (const float*)d_in[12];
  const float* W_out      = (const float*)d_in[13];
  const float* b_out      = (const float*)d_in[14];
  const float* W_copy     = (const float*)d_in[15];
  const float* b_copy     = (const float*)d_in[16];

  float* out_gen   = (float*)d_out;                       // (B, V)
  float* out_hid   = out_gen  + (size_t)Bn * Vn;          // (B, 1, H2)
  float* out_attn  = out_hid  + (size_t)Bn * H2n;         // (B, 1, T)
  float* out_copyw = out_attn + (size_t)Bn * Tn;          // (B, T)

  float* ws      = (float*)d_ws;
  float* ws_ctx  = ws;                                    // B*H2
  float* ws_emb  = ws_ctx  + Bn * H2n;                    // B*E
  float* ws_x    = ws_emb  + Bn * En;                     // B*E
  float* ws_gi   = ws_x    + Bn * En;                     // B*3H2
  float* ws_gh   = ws_gi   + Bn * G3n;                    // B*3H2
  float* ws_hn   = ws_gh   + Bn * G3n;                    // B*H2
  float* ws_feat = ws_hn   + Bn * H2n;                    // B*2304
  float* ws_log  = ws_feat + (size_t)Bn * KOUTn;          // B*V
  float* ws_attn = ws_log  + (size_t)Bn * Vn;             // B*T
  float* ws_cpl  = ws_attn + Bn * Tn;                     // B*T

  attn_context_kernel<<<Bn, 512, 0, stream>>>(enc_out, pre_hidden, W_align, b_align,
                                              out_attn, ws_attn, ws_ctx);
  comb_kernel<<<Bn, 256, 0, stream>>>(input_ids, emb_table, ws_ctx, W_comb, b_comb,
                                      ws_emb, ws_x);
  gates_kernel<<<G3n, 256, 0, stream>>>(W_ih, W_hh, b_ih, b_hh, ws_x, pre_hidden,
                                        ws_gi, ws_gh);
  gru_feat_kernel<<<Bn, 256, 0, stream>>>(ws_gi, ws_gh, pre_hidden, ws_ctx, ws_emb,
                                          out_hid, ws_hn, ws_feat);
  out_gemm_kernel<<<Vn / 16, 128, 0, stream>>>(ws_feat, W_out, b_out, ws_log);
  gen_softmax_kernel<<<Bn, 256, 0, stream>>>(ws_log, out_gen);
  pos_scatter_kernel<<<Bn, 512, 0, stream>>>(source_in, ws_attn, out_gen);
  zero_kernel<<<(Bn * Tn + 255) / 256, 256, 0, stream>>>(ws_cpl, Bn * Tn);
  copy_gemm_kernel<<<1600 * 16, 128, 0, stream>>>(enc_out, W_copy, b_copy, ws_hn, ws_cpl);
  copy_softmax_kernel<<<Bn, 512, 0, stream>>>(ws_cpl, out_copyw);
}